// HashGrid4D_48378511622633
// MI455X (gfx1250) — compile-verified
//
#include <hip/hip_runtime.h>
#include <cstdint>
#include <cmath>

typedef float v4f __attribute__((ext_vector_type(4)));

#define NLVL 8
#define TRES 8
#define HP1 2654435761u
#define HP2 805459861u

__device__ __forceinline__ v4f tload(const float* __restrict__ tbl, uint32_t idx) {
  return *reinterpret_cast<const v4f*>(tbl + ((size_t)idx << 2));
}

// 2D bilinear hash-grid lookup: 4 gathers (primes: 1, HP1)
__device__ __forceinline__ v4f enc2d(const float* __restrict__ tbl, uint32_t mask,
                                     uint32_t pa, uint32_t pb, float fa, float fb) {
  uint32_t a0 = pa, a1 = pa + 1u;
  uint32_t b0 = pb * HP1, b1 = (pb + 1u) * HP1;
  v4f v00 = tload(tbl, (a0 ^ b0) & mask);
  v4f v10 = tload(tbl, (a1 ^ b0) & mask);
  v4f v01 = tload(tbl, (a0 ^ b1) & mask);
  v4f v11 = tload(tbl, (a1 ^ b1) & mask);
  float ga = 1.0f - fa, gb = 1.0f - fb;
  v4f r = (ga * gb) * v00;
  r += (fa * gb) * v10;
  r += (ga * fb) * v01;
  r += (fa * fb) * v11;
  return r;
}

// 3D trilinear hash-grid lookup: 8 gathers (primes: 1, HP1, HP2)
__device__ __forceinline__ v4f enc3d(const float* __restrict__ tbl, uint32_t mask,
                                     uint32_t px, uint32_t py, uint32_t pz,
                                     float fx, float fy, float fz) {
  uint32_t xa = px,        xb = px + 1u;
  uint32_t ya = py * HP1,  yb = (py + 1u) * HP1;
  uint32_t za = pz * HP2,  zb = (pz + 1u) * HP2;
  v4f v000 = tload(tbl, (xa ^ ya ^ za) & mask);
  v4f v100 = tload(tbl, (xb ^ ya ^ za) & mask);
  v4f v010 = tload(tbl, (xa ^ yb ^ za) & mask);
  v4f v110 = tload(tbl, (xb ^ yb ^ za) & mask);
  v4f v001 = tload(tbl, (xa ^ ya ^ zb) & mask);
  v4f v101 = tload(tbl, (xb ^ ya ^ zb) & mask);
  v4f v011 = tload(tbl, (xa ^ yb ^ zb) & mask);
  v4f v111 = tload(tbl, (xb ^ yb ^ zb) & mask);
  float gx = 1.0f - fx, gy = 1.0f - fy, gz = 1.0f - fz;
  v4f r = (gx * gy * gz) * v000;
  r += (fx * gy * gz) * v100;
  r += (gx * fy * gz) * v010;
  r += (fx * fy * gz) * v110;
  r += (gx * gy * fz) * v001;
  r += (fx * gy * fz) * v101;
  r += (gx * fy * fz) * v011;
  r += (fx * fy * fz) * v111;
  return r;
}

// One time-blended 2D plane: 8 levels, two time slices, Lagrange-4 combine.
__device__ __forceinline__ void plane_encode(
    float* __restrict__ outp, const float* __restrict__ tbl, uint32_t mask,
    float a, float b, int i1, int i2, float w1, float w2,
    float c0, float c1, float c2, float c3, const float* __restrict__ sscale) {
  const size_t lstride = (size_t)(mask + 1u) * 4u;
  const size_t sstride = lstride * NLVL;
  const float* tb1 = tbl + (size_t)i1 * sstride;
  const float* tb2 = tbl + (size_t)i2 * sstride;
  float o[NLVL];
#pragma unroll 2
  for (int l = 0; l < NLVL; ++l) {
    const float s = sscale[l];
    float pa = __fadd_rn(__fmul_rn(a, s), 0.5f);
    float pb = __fadd_rn(__fmul_rn(b, s), 0.5f);
    float fla = floorf(pa), flb = floorf(pb);
    float fa = pa - fla, fb = pb - flb;
    uint32_t ua = (uint32_t)fla, ub = (uint32_t)flb;
    v4f f1 = enc2d(tb1 + (size_t)l * lstride, mask, ua, ub, fa, fb);
    v4f f;
    if (i1 != i2) {  // uniform branch (t is a scalar)
      v4f f2 = enc2d(tb2 + (size_t)l * lstride, mask, ua, ub, fa, fb);
      f = w1 * f1 + w2 * f2;
    } else {
      f = f1;  // w2 == 0 exactly when i1 == i2
    }
    o[l] = c0 * f.x + c1 * f.y + c2 * f.z + c3 * f.w;
  }
  v4f lo = {o[0], o[1], o[2], o[3]};
  v4f hi = {o[4], o[5], o[6], o[7]};
  __builtin_nontemporal_store(lo, (v4f*)(outp));
  __builtin_nontemporal_store(hi, (v4f*)(outp + 4));
}

__global__ void __launch_bounds__(256)
hashgrid4d_kernel(const float* __restrict__ xg, const float* __restrict__ tg,
                  const float* __restrict__ tstat, const float* __restrict__ txy,
                  const float* __restrict__ txz, const float* __restrict__ tyz,
                  float* __restrict__ out, int N,
                  uint32_t maskS, uint32_t maskXY, uint32_t maskXZ, uint32_t maskYZ) {
  __shared__ float sscale[NLVL];
  __shared__ float sx[256 * 3];
  const int lid = threadIdx.x;

  // per-level scales, double-precision exp2 exactly like numpy, cast to f32
  if (lid < NLVL)
    sscale[lid] = (float)(exp2((double)lid * (6.0 / 7.0)) * 512.0 - 1.0);

  // Stage this block's x coordinates coalesced into LDS.
  const int64_t base3 = (int64_t)blockIdx.x * (256 * 3);
  const int64_t tot3 = (int64_t)N * 3;
#if defined(__gfx1250__) && __has_builtin(__builtin_amdgcn_global_load_async_to_lds_b32)
  {
#pragma unroll
    for (int k = 0; k < 3; ++k) {
      int64_t gi = base3 + lid + k * 256;
      if (gi < tot3) {
        __builtin_amdgcn_global_load_async_to_lds_b32(
            (__attribute__((address_space(1))) int*)(xg + gi),
            (__attribute__((address_space(3))) int*)(&sx[lid + k * 256]),
            0, 0);
      }
    }
#if __has_builtin(__builtin_amdgcn_s_wait_asynccnt)
    __builtin_amdgcn_s_wait_asynccnt(0);
#else
    asm volatile("s_wait_asynccnt 0" ::: "memory");
#endif
  }
#else
#pragma unroll
  for (int k = 0; k < 3; ++k) {
    int64_t gi = base3 + lid + k * 256;
    if (gi < tot3) sx[lid + k * 256] = __builtin_nontemporal_load(xg + gi);
  }
#endif
  __syncthreads();

  const int n = blockIdx.x * 256 + lid;
  if (n >= N) return;

  const float x0 = sx[lid * 3 + 0];
  const float x1 = sx[lid * 3 + 1];
  const float x2 = sx[lid * 3 + 2];
  const float tv = tg[0];  // uniform -> scalar load

  // time-slice blend
  const float tidx = __fmul_rn(tv, (float)(TRES - 1));
  const float tfl = floorf(tidx);
  const int i1 = (int)tfl;
  const int i2 = (int)ceilf(tidx);
  const float w2 = tidx - tfl;
  const float w1 = 1.0f - w2;

  // Lagrange basis coefficients at nodes {0, 1/3, 2/3, 1}
  const float T1 = 1.0f / 3.0f, T2 = 2.0f / 3.0f;
  const float d0 = tv, d1 = tv - T1, d2 = tv - T2, d3 = tv - 1.0f;
  const float c0 = (d1 / (0.0f - T1)) * (d2 / (0.0f - T2)) * (d3 / (0.0f - 1.0f));
  const float c1 = (d0 / (T1 - 0.0f)) * (d2 / (T1 - T2)) * (d3 / (T1 - 1.0f));
  const float c2 = (d0 / (T2 - 0.0f)) * (d1 / (T2 - T1)) * (d3 / (T2 - 1.0f));
  const float c3 = (d0 / (1.0f - 0.0f)) * (d1 / (1.0f - T1)) * (d2 / (1.0f - T2));

  // ---- static 3D hash grid: [N,32], level-major ----
  {
    float* outp = out + (size_t)n * 32;
    const size_t lstride = (size_t)(maskS + 1u) * 4u;
#pragma unroll 2
    for (int l = 0; l < NLVL; ++l) {
      const float s = sscale[l];
      float pX = __fadd_rn(__fmul_rn(x0, s), 0.5f);
      float pY = __fadd_rn(__fmul_rn(x1, s), 0.5f);
      float pZ = __fadd_rn(__fmul_rn(x2, s), 0.5f);
      float flX = floorf(pX), flY = floorf(pY), flZ = floorf(pZ);
      float fx = pX - flX, fy = pY - flY, fz = pZ - flZ;
      uint32_t ux = (uint32_t)flX, uy = (uint32_t)flY, uz = (uint32_t)flZ;
      v4f acc = enc3d(tstat + (size_t)l * lstride, maskS, ux, uy, uz, fx, fy, fz);
      __builtin_nontemporal_store(acc, (v4f*)(outp + l * 4));
    }
  }

  // ---- dynamic tri-plane time features: [N,24] ----
  float* outd = out + (size_t)N * 32 + (size_t)n * 24;
  plane_encode(outd + 0,  txy, maskXY, x0, x1, i1, i2, w1, w2, c0, c1, c2, c3, sscale);
  plane_encode(outd + 8,  txz, maskXZ, x0, x2, i1, i2, w1, w2, c0, c1, c2, c3, sscale);
  plane_encode(outd + 16, tyz, maskYZ, x1, x2, i1, i2, w1, w2, c0, c1, c2, c3, sscale);
}

extern "C" void kernel_launch(void* const* d_in, const int* in_sizes, int n_in,
                              void* d_out, int out_size, void* d_ws, size_t ws_size,
                              hipStream_t stream) {
  const float* x     = (const float*)d_in[0];
  const float* t     = (const float*)d_in[1];
  const float* tstat = (const float*)d_in[2];
  const float* txy   = (const float*)d_in[3];
  const float* txz   = (const float*)d_in[4];
  const float* tyz   = (const float*)d_in[5];
  float* out = (float*)d_out;

  const int N = in_sizes[0] / 3;
  const uint32_t Ts  = (uint32_t)(in_sizes[2] / (NLVL * 4));          // 2^19
  const uint32_t Txy = (uint32_t)(in_sizes[3] / (TRES * NLVL * 4));   // 2^15
  const uint32_t Txz = (uint32_t)(in_sizes[4] / (TRES * NLVL * 4));   // 2^13
  const uint32_t Tyz = (uint32_t)(in_sizes[5] / (TRES * NLVL * 4));   // 2^13

  const int blocks = (N + 255) / 256;
  hipLaunchKernelGGL(hashgrid4d_kernel, dim3(blocks), dim3(256), 0, stream,
                     x, t, tstat, txy, txz, tyz, out, N,
                     Ts - 1u, Txy - 1u, Txz - 1u, Tyz - 1u);
}